// ExtractLearnableSlices_70377334113103
// MI455X (gfx1250) — compile-verified
//
#include <hip/hip_runtime.h>
#include <hip/hip_bf16.h>

// Problem constants (from the reference)
#define B_   64
#define C_   64
#define L_   16384
#define N_   128
#define W_   512
#define PAD_ 520   // LDS row stride in floats (513 rounded up, keeps rows apart)

typedef __attribute__((address_space(1))) int gas_int_t;  // global
typedef __attribute__((address_space(3))) int las_int_t;  // LDS

__global__ __launch_bounds__(256) void ExtractLearnableSlices_kernel(
    const float* __restrict__ x,               // [B_, C_, L_]
    const float* __restrict__ channel_params,  // [N_]
    const float* __restrict__ offset_params,   // [N_]
    float* __restrict__ out)                   // [B_, N_, W_]
{
    __shared__ float smem[2 * PAD_];

    const int tid = threadIdx.x;
    const int bn  = blockIdx.x;          // 0 .. B_*N_-1
    const int b   = bn >> 7;             // bn / N_
    const int n   = bn & (N_ - 1);       // bn % N_

    // ---- per-slice interpolation parameters (uniform across the block) ----
    const float cp = channel_params[n];
    const float op = offset_params[n];

    const float sc      = 1.0f / (1.0f + __expf(-cp));
    const float desired = sc * (float)(C_ - 1);
    int fc = (int)floorf(desired);
    if (fc < 0) fc = 0;
    if (fc > C_ - 1) fc = C_ - 1;
    int cc = fc + 1;
    if (cc > C_ - 1) cc = C_ - 1;
    const float wch = desired - (float)fc;

    const float so = 1.0f / (1.0f + __expf(-op));
    const float t0 = so * (float)(L_ - W_);
    const int   p0 = (int)floorf(t0);
    const float wt = t0 - (float)p0;

    const float* rowF = x + ((size_t)b * C_ + (size_t)fc) * L_;
    const float* rowC = x + ((size_t)b * C_ + (size_t)cc) * L_;

    // ---- stage two 513-element windows into LDS ----
#if __has_builtin(__builtin_amdgcn_global_load_async_to_lds_b32)
    // CDNA5 async DMA path: per-lane global->LDS copies, tracked by ASYNCcnt.
    for (int e = tid; e < 2 * (W_ + 1); e += 256) {
        const int r = (e >= (W_ + 1)) ? 1 : 0;
        const int k = e - r * (W_ + 1);
        int gi = p0 + k;
        if (gi > L_ - 1) gi = L_ - 1;   // matches reference's min(pf+1, L-1) clamp
        const float* src = (r ? rowC : rowF) + gi;
        __builtin_amdgcn_global_load_async_to_lds_b32(
            (gas_int_t*)(const_cast<float*>(src)),
            (las_int_t*)(&smem[r * PAD_ + k]),
            /*imm offset*/ 0, /*cpol*/ 0);
    }
#if __has_builtin(__builtin_amdgcn_s_wait_asynccnt)
    __builtin_amdgcn_s_wait_asynccnt(0);
#else
    asm volatile("s_wait_asynccnt 0" ::: "memory");
#endif
    __syncthreads();
#else
    // Fallback: staged through VGPRs with plain loads.
    for (int e = tid; e < 2 * (W_ + 1); e += 256) {
        const int r = (e >= (W_ + 1)) ? 1 : 0;
        const int k = e - r * (W_ + 1);
        int gi = p0 + k;
        if (gi > L_ - 1) gi = L_ - 1;
        smem[r * PAD_ + k] = (r ? rowC : rowF)[gi];
    }
    __syncthreads();
#endif

    // ---- bilinear (channel, time) interpolation from LDS ----
    float* outp = out + (size_t)bn * W_;
#pragma unroll
    for (int jj = 0; jj < 2; ++jj) {
        const int   j  = tid + jj * 256;
        const float f0 = smem[j];
        const float f1 = smem[j + 1];
        const float c0 = smem[PAD_ + j];
        const float c1 = smem[PAD_ + j + 1];
        // x_channel at pf and pc, then time lerp (same algebra as reference)
        const float a0 = f0 + wch * (c0 - f0);
        const float a1 = f1 + wch * (c1 - f1);
        outp[j] = a0 + wt * (a1 - a0);
    }
}

extern "C" void kernel_launch(void* const* d_in, const int* in_sizes, int n_in,
                              void* d_out, int out_size, void* d_ws, size_t ws_size,
                              hipStream_t stream) {
    const float* x    = (const float*)d_in[0];
    const float* cpar = (const float*)d_in[1];
    const float* opar = (const float*)d_in[2];
    float*       out  = (float*)d_out;

    dim3 grid(B_ * N_);   // one workgroup per (batch, slice) pair: 8192 blocks
    dim3 block(256);      // 8 waves (wave32)
    ExtractLearnableSlices_kernel<<<grid, block, 0, stream>>>(x, cpar, opar, out);
}